// MambaIndexer_52209622450601
// MI455X (gfx1250) — compile-verified
//
#include <hip/hip_runtime.h>
#include <math.h>

#define SEQ     2048
#define DMODEL  1024
#define DINNER  1024
#define DSTATE  64
#define HEADDIM 32
#define NHEADS  32
#define CONVDIM 1152            // DINNER + 2*DSTATE
#define DINPROJ 2208            // 2*DINNER + 2*DSTATE + NHEADS
#define IDXDIM  64
#define NCHUNKS 8
#define CLEN    256             // SEQ / NCHUNKS

#define APAD    20              // padded LDS row stride for A (floats)
#define BPAD    34              // padded LDS row stride for B (floats)
#define ASTAGE  (128 * APAD)    // one A stage (floats): 128 rows x 16 k
#define BSTAGE  (16 * BPAD)     // one B stage (floats): 16 k x 32 n

typedef __attribute__((ext_vector_type(2))) float v2f;
typedef __attribute__((ext_vector_type(8))) float v8f;

__device__ __forceinline__ float silu_f(float x) { return x / (1.0f + expf(-x)); }

// ---- CDNA5 async global->LDS copies (ASYNCcnt path) -----------------------
__device__ __forceinline__ void async_copy_b128(unsigned lds_off, const float* g)
{
    asm volatile("global_load_async_to_lds_b128 %0, %1, off"
                 :: "v"(lds_off), "v"(g) : "memory");
}
__device__ __forceinline__ void async_copy_b64(unsigned lds_off, const float* g)
{
    asm volatile("global_load_async_to_lds_b64 %0, %1, off"
                 :: "v"(lds_off), "v"(g) : "memory");
}
__device__ __forceinline__ void wait_async_le3() { asm volatile("s_wait_asynccnt 0x3" ::: "memory"); }
__device__ __forceinline__ void wait_async_le0() { asm volatile("s_wait_asynccnt 0x0" ::: "memory"); }

// ---------------------------------------------------------------------------
// NN GEMM, fp32 WMMA 16x16x4, LDS-staged, software-pipelined (2-stage unroll
// so buffer offsets are immediates). Block = 256 threads (8 waves) computes a
// 128(M) x 32(N) tile; each wave owns 32x16 (two accumulators sharing B).
// Requires M%128==0, N%32==0, K%32==0 (true for all call sites).
// ---------------------------------------------------------------------------
__global__ __launch_bounds__(256)
void gemm_nn_wmma(const float* __restrict__ A, const float* __restrict__ B,
                  float* __restrict__ C, int M, int N, int K)
{
    __shared__ float shA[2 * ASTAGE];
    __shared__ float shB[2 * BSTAGE];

    const int tid   = threadIdx.x;
    const int lane  = tid & 31;
    const int wave  = tid >> 5;
    const int tileM = blockIdx.y * 128;
    const int tileN = blockIdx.x * 32;
    const int mloc  = (wave >> 1) * 32;     // wave's M offset inside tile
    const int nloc  = (wave & 1) * 16;      // wave's N offset inside tile
    const int fidx  = lane & 15;            // M index (A) / N index (B/C/D)
    const int half  = lane >> 4;
    const int koff  = half * 2;

    // cooperative load coordinates
    const int arow = tid >> 1,  acol = (tid & 1) * 8;    // A: 128 x 16, 2xfloat4/thread
    const int brow = tid >> 4,  bcol = (tid & 15) * 2;   // B: 16 x 32, float2/thread

    // LDS byte offsets (generic shared ptr truncates to LDS offset)
    const unsigned ldsA = (unsigned)(uintptr_t)&shA[arow * APAD + acol];
    const unsigned ldsB = (unsigned)(uintptr_t)&shB[brow * BPAD + bcol];
    const unsigned aBytes = ASTAGE * 4, bBytes = BSTAGE * 4;

    const float* gA = A + (size_t)(tileM + arow) * K + acol;
    const float* gB = B + (size_t)brow * N + tileN + bcol;
    const size_t  gBstep = (size_t)16 * N;

    v8f acc0 = {}, acc1 = {};

    auto issue_stage = [&](int k0, int which) {
        const unsigned ab = ldsA + (unsigned)which * aBytes;
        async_copy_b128(ab,      gA + k0);
        async_copy_b128(ab + 16, gA + k0 + 4);
        async_copy_b64 (ldsB + (unsigned)which * bBytes,
                        gB + (size_t)(k0 >> 4) * gBstep);
    };
    auto stage_compute = [&](const float* sAb, const float* sBb) {
        const float* sA0 = sAb + (mloc + fidx) * APAD;
        const float* sA1 = sA0 + 16 * APAD;
        const float* sB  = sBb + nloc + fidx;
#pragma unroll
        for (int kk = 0; kk < 4; ++kk) {
            const int k = kk * 4 + koff;
            v2f a0, a1, b;
            a0.x = sA0[k];        a0.y = sA0[k + 1];
            a1.x = sA1[k];        a1.y = sA1[k + 1];
            b.x  = sB[k * BPAD];  b.y  = sB[(k + 1) * BPAD];
            acc0 = __builtin_amdgcn_wmma_f32_16x16x4_f32(
                false, a0, false, b, (short)0, acc0, false, false);
            acc1 = __builtin_amdgcn_wmma_f32_16x16x4_f32(
                false, a1, false, b, (short)0, acc1, false, false);
        }
    };

    issue_stage(0, 0);                       // stage 0 in flight
    for (int k0 = 0; k0 < K; k0 += 32) {
        issue_stage(k0 + 16, 1);             // next stage -> buf1
        wait_async_le3();                    // buf0's 3 copies done
        __syncthreads();
        stage_compute(shA, shB);             // consume buf0
        __syncthreads();

        if (k0 + 32 < K) { issue_stage(k0 + 32, 0); wait_async_le3(); }
        else             { wait_async_le0(); }
        __syncthreads();
        stage_compute(shA + ASTAGE, shB + BSTAGE);   // consume buf1
        __syncthreads();
    }

#pragma unroll
    for (int j = 0; j < 8; ++j) {
        int m = tileM + mloc + j + half * 8;     // C/D: VGPR j holds rows j, j+8
        size_t col = (size_t)tileN + nloc + fidx;
        C[(size_t)m * N + col]        = acc0[j];
        C[(size_t)(m + 16) * N + col] = acc1[j];
    }
}

// ---------------------------------------------------------------------------
// Causal-masked NT GEMM: O[t,s] = scale * sum_i Q[t,i]*Km[s,i], -inf for s>t.
// K = 64 here -> fully unrolled, addresses fold to immediate offsets.
// ---------------------------------------------------------------------------
__global__ __launch_bounds__(128)
void gemm_nt_causal(const float* __restrict__ Q, const float* __restrict__ Km,
                    float* __restrict__ O, int M, int N, float scale)
{
    const int lane  = threadIdx.x & 31;
    const int wave  = threadIdx.x >> 5;
    const int tileM = blockIdx.y * 32 + (wave >> 1) * 16;
    const int tileN = blockIdx.x * 32 + (wave & 1) * 16;
    const int fidx  = lane & 15;
    const int half  = lane >> 4;
    const int koff  = half * 2;

    v8f acc = {};
    const float* Arow = Q  + (size_t)(tileM + fidx) * IDXDIM + koff;
    const float* Brow = Km + (size_t)(tileN + fidx) * IDXDIM + koff;
#pragma unroll
    for (int k0 = 0; k0 < IDXDIM; k0 += 4) {
        v2f a, b;
        a.x = Arow[k0];
        a.y = Arow[k0 + 1];
        b.x = Brow[k0];
        b.y = Brow[k0 + 1];
        acc = __builtin_amdgcn_wmma_f32_16x16x4_f32(
            false, a, false, b, (short)0, acc, false, false);
    }
    const float ninf = -__builtin_inff();
#pragma unroll
    for (int j = 0; j < 8; ++j) {
        int m = tileM + j + half * 8;           // query index t
        int n = tileN + fidx;                   // key index s
        float v = acc[j] * scale;
        O[(size_t)m * N + n] = (n <= m) ? v : ninf;
    }
}

// ---------------------------------------------------------------------------
// dt = softplus(proj + bias); dA = exp(dt * -exp(A_log))
// ---------------------------------------------------------------------------
__global__ void dt_kernel(const float* __restrict__ zxbcdt,
                          const float* __restrict__ dt_bias,
                          const float* __restrict__ A_log,
                          float* __restrict__ dtb, float* __restrict__ dAb)
{
    int idx = blockIdx.x * blockDim.x + threadIdx.x;
    if (idx >= SEQ * NHEADS) return;
    int t = idx / NHEADS, h = idx % NHEADS;
    float v  = zxbcdt[(size_t)t * DINPROJ + (DINNER + CONVDIM) + h] + dt_bias[h];
    float dt = (v > 20.0f) ? v : log1pf(expf(v));
    float A  = -expf(A_log[h]);
    dtb[idx] = dt;
    dAb[idx] = expf(dt * A);
}

// ---------------------------------------------------------------------------
// Depthwise causal conv (k=4) + bias + SiLU over the xBC slice of zxbcdt.
// ---------------------------------------------------------------------------
__global__ void conv_silu_kernel(const float* __restrict__ zxbcdt,
                                 const float* __restrict__ conv_w,
                                 const float* __restrict__ conv_b,
                                 float* __restrict__ xBC)
{
    int idx = blockIdx.x * blockDim.x + threadIdx.x;
    if (idx >= SEQ * CONVDIM) return;
    int t = idx / CONVDIM, c = idx % CONVDIM;
    float s = 0.0f;
#pragma unroll
    for (int j = 0; j < 4; ++j) {
        int tt = t - 3 + j;
        if (tt >= 0)
            s += zxbcdt[(size_t)tt * DINPROJ + DINNER + c] * conv_w[c * 4 + j];
    }
    s += conv_b[c];
    xBC[(size_t)t * CONVDIM + c] = silu_f(s);
}

// ---------------------------------------------------------------------------
// Chunk-local scan. Grid (NHEADS, NCHUNKS), 512 threads: thread (p,ng) holds
// state h[p, ng*4..ng*4+3] in registers. Emits y_local + D*xs (ungated) and
// the chunk-final state Sloc.
// ---------------------------------------------------------------------------
__global__ __launch_bounds__(512)
void scan_local_kernel(const float* __restrict__ xBC, const float* __restrict__ dtb,
                       const float* __restrict__ dAb, const float* __restrict__ Dp,
                       float* __restrict__ ybuf, float* __restrict__ Sloc)
{
    const int h  = blockIdx.x, c = blockIdx.y;
    const int tid = threadIdx.x;
    const int p  = tid >> 4, ng = tid & 15, n0 = ng * 4;
    __shared__ float Bs[DSTATE], Cs[DSTATE], xss[HEADDIM];
    float s0 = 0, s1 = 0, s2 = 0, s3 = 0;
    const float Dh = Dp[h];
    const int t0 = c * CLEN;

    for (int i = 0; i < CLEN; ++i) {
        int t = t0 + i;
        const float* row = xBC + (size_t)t * CONVDIM;
        if (tid < DSTATE)                { Bs[tid] = row[DINNER + tid];
                                           Cs[tid] = row[DINNER + DSTATE + tid]; }
        else if (tid < DSTATE + HEADDIM) { xss[tid - DSTATE] = row[h * HEADDIM + tid - DSTATE]; }
        __syncthreads();
        float dA = dAb[(size_t)t * NHEADS + h];
        float co = dtb[(size_t)t * NHEADS + h] * xss[p];
        s0 = dA * s0 + co * Bs[n0 + 0];
        s1 = dA * s1 + co * Bs[n0 + 1];
        s2 = dA * s2 + co * Bs[n0 + 2];
        s3 = dA * s3 + co * Bs[n0 + 3];
        float part = s0 * Cs[n0 + 0] + s1 * Cs[n0 + 1]
                   + s2 * Cs[n0 + 2] + s3 * Cs[n0 + 3];
        part += __shfl_xor(part, 8, 16);
        part += __shfl_xor(part, 4, 16);
        part += __shfl_xor(part, 2, 16);
        part += __shfl_xor(part, 1, 16);
        if (ng == 0)
            ybuf[(size_t)t * DINNER + h * HEADDIM + p] = part + Dh * xss[p];
        __syncthreads();
    }
    size_t base = (((size_t)h * NCHUNKS + c) * HEADDIM + p) * DSTATE + n0;
    Sloc[base + 0] = s0; Sloc[base + 1] = s1;
    Sloc[base + 2] = s2; Sloc[base + 3] = s3;
}

// ---------------------------------------------------------------------------
// Cross-chunk combine: alpha[c] = prod dA over chunk (scalar per head),
// Hpre[c] = alpha[c-1]*Hpre[c-1] + Sloc[c-1]; also in-chunk inclusive
// cumulative decay cumdA[t,h]. Grid = NHEADS, 512 threads (4 state elems each).
// ---------------------------------------------------------------------------
__global__ __launch_bounds__(512)
void combine_kernel(const float* __restrict__ dAb, const float* __restrict__ Sloc,
                    float* __restrict__ Hpre, float* __restrict__ cumdA)
{
    const int h = blockIdx.x, tid = threadIdx.x;
    __shared__ float alpha[NCHUNKS];
    if (tid < NCHUNKS) {
        float cum = 1.0f;
        for (int i = 0; i < CLEN; ++i) {
            size_t ti = (size_t)(tid * CLEN + i) * NHEADS + h;
            cum *= dAb[ti];
            cumdA[ti] = cum;
        }
        alpha[tid] = cum;
    }
    __syncthreads();
    const int e0 = tid * 4;
    float h0 = 0, h1 = 0, h2 = 0, h3 = 0;
    for (int c = 0; c < NCHUNKS; ++c) {
        size_t base = ((size_t)h * NCHUNKS + c) * (HEADDIM * DSTATE) + e0;
        Hpre[base + 0] = h0; Hpre[base + 1] = h1;
        Hpre[base + 2] = h2; Hpre[base + 3] = h3;
        float a = alpha[c];
        h0 = a * h0 + Sloc[base + 0];
        h1 = a * h1 + Sloc[base + 1];
        h2 = a * h2 + Sloc[base + 2];
        h3 = a * h3 + Sloc[base + 3];
    }
}

// ---------------------------------------------------------------------------
// y[t,hp] += cumdA(t) * (C_t . Hpre[chunk(t)]);  then gate with silu(z).
// ---------------------------------------------------------------------------
__global__ void correct_gate_kernel(const float* __restrict__ xBC,
                                    const float* __restrict__ zxbcdt,
                                    const float* __restrict__ Hpre,
                                    const float* __restrict__ cumdA,
                                    float* __restrict__ ybuf)
{
    int idx = blockIdx.x * blockDim.x + threadIdx.x;
    if (idx >= SEQ * DINNER) return;
    int t = idx >> 10, hp = idx & 1023;
    int h = hp >> 5, p = hp & 31;
    int c = t >> 8;  // CLEN = 256
    const float* Hp   = Hpre + (((size_t)h * NCHUNKS + c) * HEADDIM + p) * DSTATE;
    const float* Crow = xBC + (size_t)t * CONVDIM + DINNER + DSTATE;
    float s = 0.0f;
#pragma unroll
    for (int n = 0; n < DSTATE; ++n) s += Crow[n] * Hp[n];
    float y = ybuf[idx] + cumdA[(size_t)t * NHEADS + h] * s;
    float z = zxbcdt[(size_t)t * DINPROJ + hp];
    ybuf[idx] = y * silu_f(z);
}

// ---------------------------------------------------------------------------
// In-place RMS norm over each row of 1024, * norm_w.
// ---------------------------------------------------------------------------
__global__ __launch_bounds__(256)
void rmsnorm_kernel(float* __restrict__ y, const float* __restrict__ w)
{
    const int t = blockIdx.x, tid = threadIdx.x;
    __shared__ float red[256];
    float* row = y + (size_t)t * DINNER;
    float v0 = row[tid], v1 = row[tid + 256], v2 = row[tid + 512], v3 = row[tid + 768];
    red[tid] = v0 * v0 + v1 * v1 + v2 * v2 + v3 * v3;
    __syncthreads();
    for (int off = 128; off > 0; off >>= 1) {
        if (tid < off) red[tid] += red[tid + off];
        __syncthreads();
    }
    float r = rsqrtf(red[0] * (1.0f / DINNER) + 1e-5f);
    row[tid      ] = v0 * r * w[tid];
    row[tid + 256] = v1 * r * w[tid + 256];
    row[tid + 512] = v2 * r * w[tid + 512];
    row[tid + 768] = v3 * r * w[tid + 768];
}

// ---------------------------------------------------------------------------
extern "C" void kernel_launch(void* const* d_in, const int* in_sizes, int n_in,
                              void* d_out, int out_size, void* d_ws, size_t ws_size,
                              hipStream_t stream)
{
    (void)in_sizes; (void)n_in; (void)out_size; (void)ws_size;
    const float* x       = (const float*)d_in[0];
    const float* W_in    = (const float*)d_in[1];
    const float* conv_w  = (const float*)d_in[2];
    const float* conv_b  = (const float*)d_in[3];
    const float* dt_bias = (const float*)d_in[4];
    const float* A_log   = (const float*)d_in[5];
    const float* Dp      = (const float*)d_in[6];
    const float* norm_w  = (const float*)d_in[7];
    const float* W_out   = (const float*)d_in[8];
    const float* Wq      = (const float*)d_in[9];
    const float* Wk      = (const float*)d_in[10];
    float* out = (float*)d_out;

    // workspace layout (floats)
    float* wsf    = (float*)d_ws;
    float* zxbcdt = wsf;                                  // 2048*2208
    float* xBC    = zxbcdt + (size_t)SEQ * DINPROJ;       // 2048*1152
    float* dtb    = xBC    + (size_t)SEQ * CONVDIM;       // 2048*32
    float* dAb    = dtb    + (size_t)SEQ * NHEADS;        // 2048*32
    float* cumdA  = dAb    + (size_t)SEQ * NHEADS;        // 2048*32
    float* Sloc   = cumdA  + (size_t)SEQ * NHEADS;        // 32*8*32*64
    float* Hpre   = Sloc   + (size_t)NHEADS * NCHUNKS * HEADDIM * DSTATE;
    float* ybuf   = Hpre   + (size_t)NHEADS * NCHUNKS * HEADDIM * DSTATE; // 2048*1024
    float* q      = ybuf   + (size_t)SEQ * DINNER;        // 2048*64
    float* kk     = q      + (size_t)SEQ * IDXDIM;        // 2048*64
    float* xmamba = zxbcdt;  // alias: zxbcdt is dead after correct_gate

    // 1) in-projection: zxbcdt = x @ W_in   (2048x1024x2208, fp32 WMMA)
    gemm_nn_wmma<<<dim3(DINPROJ / 32, SEQ / 128), 256, 0, stream>>>(
        x, W_in, zxbcdt, SEQ, DINPROJ, DMODEL);
    // 2) dt / dA
    dt_kernel<<<(SEQ * NHEADS + 255) / 256, 256, 0, stream>>>(
        zxbcdt, dt_bias, A_log, dtb, dAb);
    // 3) depthwise conv + SiLU
    conv_silu_kernel<<<(SEQ * CONVDIM + 255) / 256, 256, 0, stream>>>(
        zxbcdt, conv_w, conv_b, xBC);
    // 4) chunked selective scan
    scan_local_kernel<<<dim3(NHEADS, NCHUNKS), 512, 0, stream>>>(
        xBC, dtb, dAb, Dp, ybuf, Sloc);
    combine_kernel<<<NHEADS, 512, 0, stream>>>(dAb, Sloc, Hpre, cumdA);
    correct_gate_kernel<<<(SEQ * DINNER + 255) / 256, 256, 0, stream>>>(
        xBC, zxbcdt, Hpre, cumdA, ybuf);
    // 5) RMS norm (in place)
    rmsnorm_kernel<<<SEQ, 256, 0, stream>>>(ybuf, norm_w);
    // 6) out-projection + q/k projections (fp32 WMMA)
    gemm_nn_wmma<<<dim3(DMODEL / 32, SEQ / 128), 256, 0, stream>>>(
        ybuf, W_out, xmamba, SEQ, DMODEL, DINNER);
    gemm_nn_wmma<<<dim3(IDXDIM / 32, SEQ / 128), 256, 0, stream>>>(
        xmamba, Wq, q, SEQ, IDXDIM, DMODEL);
    gemm_nn_wmma<<<dim3(IDXDIM / 32, SEQ / 128), 256, 0, stream>>>(
        x, Wk, kk, SEQ, IDXDIM, DMODEL);
    // 7) causal masked scores = (q @ k^T) / 8
    gemm_nt_causal<<<dim3(SEQ / 32, SEQ / 32), 128, 0, stream>>>(
        q, kk, out, SEQ, SEQ, 0.125f);
}